// NeuralNDCG_3435973837343
// MI455X (gfx1250) — compile-verified
//
#include <hip/hip_runtime.h>
#include <hip/hip_bf16.h>
#include <climits>

// ---------------------------------------------------------------------------
// NeuralNDCG for MI455X (gfx1250, wave32, WGP with 320KB LDS).
// One workgroup per batch row; the 256x256 Sinkhorn matrix lives entirely in
// LDS (row stride padded to 260 floats for bank-conflict-free column passes).
// Row/col sums and the final matvec run on the matrix pipe via
// V_WMMA_F32_16X16X4_F32 (matrix x ones / broadcast-vector), co-executing
// with the VALU normalization divides.
// ---------------------------------------------------------------------------

#define NN        256
#define BATCH     128
#define STRIDE    260                     // 256 + 4 pad: conflict-free passes
#define NEG_BIG   (-100000000.0f)
#define PADV      (-1.0f)
#define SINK_ITERS 50

// LDS layout (floats) inside one dynamic blob
#define P_OFF     0
#define SNEG_OFF  (NN * STRIDE)           // 66560
#define S0_OFF    (SNEG_OFF + NN)
#define GAIN_OFF  (S0_OFF + NN)
#define SCAL_OFF  (GAIN_OFF + NN)
#define BM_OFF    (SCAL_OFF + NN)
#define MASK_OFF  (BM_OFF + NN)
#define RED_OFF   (MASK_OFF + NN)
#define HIST_OFF  (RED_OFF + NN)          // 256 ints (reinterpreted)
#define MISC_OFF  (HIST_OFF + NN)         // a few scalars
#define SMEM_FLOATS (MISC_OFF + 16)       // 68624 floats = 274,496 B < 320KB

typedef __attribute__((ext_vector_type(2))) float v2f;
typedef __attribute__((ext_vector_type(8))) float v8f;

__device__ __forceinline__ v8f wmma_f32x4(v2f a, v2f b, v8f c) {
  // D(16x16) = A(16x4) * B(4x16) + C
  return __builtin_amdgcn_wmma_f32_16x16x4_f32(false, a, false, b,
                                               (short)0, c, false, false);
}

// ---------------------------------------------------------------------------
// Kernel 0: global max/min of y_true (reference uses yt.max(), yt.min()).
// ---------------------------------------------------------------------------
__global__ __launch_bounds__(256) void ndcg_minmax(const int* __restrict__ yt,
                                                   int n, int* __restrict__ mm) {
  __shared__ int smax[256];
  __shared__ int smin[256];
  const int t = threadIdx.x;
  int mx = INT_MIN, mn = INT_MAX;
  for (int i = t; i < n; i += 256) {
    const int v = yt[i];
    mx = max(mx, v);
    mn = min(mn, v);
  }
  smax[t] = mx; smin[t] = mn;
  __syncthreads();
  for (int off = 128; off > 0; off >>= 1) {
    if (t < off) {
      smax[t] = max(smax[t], smax[t + off]);
      smin[t] = min(smin[t], smin[t + off]);
    }
    __syncthreads();
  }
  if (t == 0) { mm[0] = smax[0]; mm[1] = smin[0]; }
}

// ---------------------------------------------------------------------------
// Kernel 1: per-batch pipeline, one workgroup (256 threads = 8 wave32) per b.
// ---------------------------------------------------------------------------
__global__ __launch_bounds__(256, 1)
void ndcg_main(const float* __restrict__ y_pred, const int* __restrict__ y_true,
               float* __restrict__ ndcg_out, float* __restrict__ idcg_out,
               const int* __restrict__ mm) {
  extern __shared__ float sm[];
  float* P     = sm + P_OFF;
  float* sneg  = sm + SNEG_OFF;
  float* s0    = sm + S0_OFF;
  float* gains = sm + GAIN_OFF;
  float* scal  = sm + SCAL_OFF;
  float* bm    = sm + BM_OFF;
  float* maskv = sm + MASK_OFF;
  float* red   = sm + RED_OFF;
  int*   hist  = (int*)(sm + HIST_OFF);
  int*   misc  = (int*)(sm + MISC_OFF);

  const int b    = blockIdx.x;
  const int t    = threadIdx.x;
  const int lane = t & 31;
  const int wave = t >> 5;
  const int l15  = lane & 15;
  const int half = (lane < 16) ? 0 : 2;   // K-pair selector per ISA A/B layouts

  const int maxv = mm[0];
  const int minv = mm[1];

  // ---- Phase A: per-element prep ----------------------------------------
  const float pv   = y_pred[b * NN + t];
  const int   yti  = maxv - y_true[b * NN + t] + minv;   // relevancy flip
  const float ytv  = (float)yti;
  const bool  mk   = (ytv == PADV);
  sneg[t]  = mk ? NEG_BIG : pv;
  s0[t]    = mk ? 0.0f : pv;
  gains[t] = mk ? 0.0f : (exp2f(ytv) - 1.0f);
  maskv[t] = mk ? 1.0f : 0.0f;
  hist[t]  = 0;
  if (t == 0) misc[0] = 0;
  __syncthreads();
  if (mk) atomicAdd(&misc[0], 1);
  else {
    const int hidx = yti - minv;
    if (hidx >= 0 && hidx < NN) atomicAdd(&hist[hidx], 1);
  }
  __syncthreads();
  const int   m  = misc[0];
  const float nm = (float)(NN - m);
  scal[t] = (t < NN - m) ? (nm + 1.0f - 2.0f * (float)(t + 1)) : 0.0f;
  __syncthreads();

  // ---- Phase B: B_mat[i] = sum_j (mor ? 0 : |sneg_i - sneg_j|) -----------
  const float mi = maskv[t];
  {
    const float si = sneg[t];
    float acc = 0.0f;
    for (int j = 0; j < NN; ++j) {                 // LDS broadcast reads
      const float morf = mi + maskv[j];
      const float v = fabsf(si - sneg[j]);
      acc += (morf > 0.0f) ? 0.0f : v;
    }
    bm[t] = acc;
  }
  __syncthreads();

  // ---- Phase C: P_max row t + softmax + Sinkhorn pre-mask ----------------
  {
    const float scali = scal[t];
    float rowmax = -INFINITY;
    for (int j = 0; j < NN; ++j) {
      const float morf  = mi + maskv[j];
      const float mandf = mi * maskv[j];
      float v = s0[j] * scali - bm[j];
      v = (morf  > 0.0f) ? -INFINITY : v;
      v = (mandf > 0.0f) ? 1.0f : v;
      P[t * STRIDE + j] = v;
      rowmax = fmaxf(rowmax, v);
    }
    float ssum = 0.0f;
    for (int j = 0; j < NN; ++j) {
      const float e = __expf(P[t * STRIDE + j] - rowmax);   // tau == 1
      P[t * STRIDE + j] = e;
      ssum += e;
    }
    const float inv = 1.0f / ssum;
    for (int j = 0; j < NN; ++j) {
      const float morf  = mi + maskv[j];
      const float mandf = mi * maskv[j];
      float v = P[t * STRIDE + j] * inv;
      v = (morf  > 0.0f) ? 0.0f : v;
      v = (mandf > 0.0f) ? 1.0f : v;
      P[t * STRIDE + j] = v;
    }
  }
  __syncthreads();

  // ---- Phase D: 50 Sinkhorn iterations; sums on matrix pipe (WMMA) -------
  const v2f ones = {1.0f, 1.0f};
  for (int it = 0; it < SINK_ITERS; ++it) {
    // column sums: D = Ones(16x4) x P(4x16); wave w owns column tiles 2w,2w+1
    for (int tile = wave * 2; tile < wave * 2 + 2; ++tile) {
      const int c0 = tile * 16;
      v8f acc = {};
#pragma unroll 4
      for (int k0 = 0; k0 < NN; k0 += 4) {
        v2f bb;                                   // B 4x16 tile per ISA layout
        bb.x = P[(k0 + half) * STRIDE + c0 + l15];
        bb.y = P[(k0 + half + 1) * STRIDE + c0 + l15];
        acc = wmma_f32x4(ones, bb, acc);
      }
      if (lane < 16) red[c0 + l15] = acc[0];      // row 0 of D = colsums
    }
    __syncthreads();
    // column scale: lanes hit consecutive addresses -> conflict-free
    {
      const float inv = 1.0f / fmaxf(red[t], 1e-10f);
      for (int i = 0; i < NN; ++i) P[i * STRIDE + t] *= inv;
    }
    __syncthreads();
    // row sums: D = P(16x4) x Ones(4x16); wave w owns row stripes 2w,2w+1
    for (int stripe = wave * 2; stripe < wave * 2 + 2; ++stripe) {
      const int r0 = stripe * 16;
      v8f acc = {};
#pragma unroll 4
      for (int k0 = 0; k0 < NN; k0 += 4) {
        const v2f a = *(const v2f*)&P[(r0 + l15) * STRIDE + k0 + half];
        acc = wmma_f32x4(a, ones, acc);
      }
      if (lane == 0 || lane == 16) {
        const int base = r0 + ((lane == 16) ? 8 : 0);
#pragma unroll
        for (int r = 0; r < 8; ++r) red[base + r] = acc[r];
      }
    }
    __syncthreads();
    red[t] = 1.0f / fmaxf(red[t], 1e-10f);        // invert once per row
    __syncthreads();
    for (int i = 0; i < NN; ++i) P[i * STRIDE + t] *= red[i];
    __syncthreads();
  }

  // ---- Phase E: ground_truth = P x gains via WMMA (broadcast-B) ----------
  for (int stripe = wave * 2; stripe < wave * 2 + 2; ++stripe) {
    const int r0 = stripe * 16;
    v8f acc = {};
#pragma unroll 4
    for (int k0 = 0; k0 < NN; k0 += 4) {
      const v2f a = *(const v2f*)&P[(r0 + l15) * STRIDE + k0 + half];
      v2f g;
      g.x = gains[k0 + half];
      g.y = gains[k0 + half + 1];
      acc = wmma_f32x4(a, g, acc);
    }
    if (lane == 0 || lane == 16) {
      const int base = r0 + ((lane == 16) ? 8 : 0);
#pragma unroll
      for (int r = 0; r < 8; ++r) red[base + r] = acc[r];
    }
  }
  __syncthreads();

  // discounted gains; mor row-mask (col mask already in gains==0)
  const float disc = 1.0f / log2f((float)t + 2.0f);
  const float val  = red[t] * disc * (1.0f - mi);
  __syncthreads();
  red[t] = val;
  __syncthreads();
  for (int off = 128; off > 0; off >>= 1) {
    if (t < off) red[t] += red[t + off];
    __syncthreads();
  }

  // IDCG via counting sort over the (small) integer relevancy range
  if (t == 0) {
    float idcg = 0.0f;
    int   pos  = 0;
    int   vtop = maxv - minv;
    if (vtop > NN - 1) vtop = NN - 1;
    for (int v = vtop; v >= 0; --v) {
      const int   c = hist[v];
      const float g = exp2f((float)(minv + v)) - 1.0f;
      for (int q = 0; q < c; ++q) {
        idcg += g / log2f((float)pos + 2.0f);
        ++pos;
      }
    }
    const float dg = red[0];
    ndcg_out[b] = (idcg == 0.0f) ? 0.0f : dg / (idcg + 1e-10f);
    idcg_out[b] = idcg;
  }
}

// ---------------------------------------------------------------------------
// Kernel 2: mean over batches with nonzero IDCG; negate.
// ---------------------------------------------------------------------------
__global__ __launch_bounds__(128) void ndcg_finalize(const float* __restrict__ ndcg,
                                                     const float* __restrict__ idcg,
                                                     float* __restrict__ out) {
  __shared__ float ssum[128];
  __shared__ int   scnt[128];
  const int t = threadIdx.x;
  ssum[t] = ndcg[t];
  scnt[t] = (idcg[t] != 0.0f) ? 1 : 0;
  __syncthreads();
  for (int off = 64; off > 0; off >>= 1) {
    if (t < off) { ssum[t] += ssum[t + off]; scnt[t] += scnt[t + off]; }
    __syncthreads();
  }
  if (t == 0) {
    const int   cnt = scnt[0];
    const float den = fmaxf((float)cnt, 1.0f);
    out[0] = (cnt > 0) ? -(ssum[0] / den) : 0.0f;
  }
}

// ---------------------------------------------------------------------------
extern "C" void kernel_launch(void* const* d_in, const int* in_sizes, int n_in,
                              void* d_out, int out_size, void* d_ws, size_t ws_size,
                              hipStream_t stream) {
  const float* y_pred = (const float*)d_in[0];
  const int*   y_true = (const int*)d_in[1];
  float*       out    = (float*)d_out;

  int*   mm    = (int*)d_ws;               // [0]=max, [1]=min
  float* wsf   = (float*)d_ws;
  float* ndcgb = wsf + 2;                  // 128 floats
  float* idcgb = wsf + 2 + BATCH;          // 128 floats

  const int ntot = in_sizes[1];            // 128*256

  ndcg_minmax<<<1, 256, 0, stream>>>(y_true, ntot, mm);

  const size_t shbytes = (size_t)SMEM_FLOATS * sizeof(float);  // ~274 KB
  ndcg_main<<<BATCH, 256, shbytes, stream>>>(y_pred, y_true, ndcgb, idcgb, mm);

  ndcg_finalize<<<1, 128, 0, stream>>>(ndcgb, idcgb, out);
}